// RegionProposal_56461640073891
// MI455X (gfx1250) — compile-verified
//
#include <hip/hip_runtime.h>
#include <stdint.h>

// ---------------- problem constants (match reference) ----------------
#define NUM_ANCHORS 9
#define FEAT_H 128
#define FEAT_W 128
#define NTOT (FEAT_H * FEAT_W * NUM_ANCHORS) // 147456
#define PRE_NMS 6000
#define POST_NMS 300
#define IOU_THR 0.7f

typedef uint32_t u32;
typedef unsigned int uint;
typedef uint  v4u __attribute__((ext_vector_type(4)));
typedef int   v8i __attribute__((ext_vector_type(8)));
typedef int   v4i __attribute__((ext_vector_type(4)));
typedef float v2f __attribute__((ext_vector_type(2)));
typedef float v8f __attribute__((ext_vector_type(8)));

#define NEG_INF (-__builtin_inff())

// Monotone float -> uint key (order preserving), and inverse.
__device__ __forceinline__ u32 flipKey(float f) {
    u32 b = __float_as_uint(f);
    return (b & 0x80000000u) ? ~b : (b | 0x80000000u);
}
__device__ __forceinline__ float unflipKey(u32 k) {
    u32 b = (k & 0x80000000u) ? (k & 0x7FFFFFFFu) : ~k;
    return __uint_as_float(b);
}

// ---------------------------------------------------------------------
// K1: foreground score per anchor.  softmax over 2 classes, take class 1
//     == sigmoid(s1 - s0).  Store order-preserving uint key.
// ---------------------------------------------------------------------
__global__ void rpn_score_kernel(const float* __restrict__ scores,
                                 u32* __restrict__ keys) {
    int i = blockIdx.x * blockDim.x + threadIdx.x;
    if (i >= NTOT) return;
    int p = i / NUM_ANCHORS;
    int a = i - p * NUM_ANCHORS;
    float s0 = scores[p * (2 * NUM_ANCHORS) + a];
    float s1 = scores[p * (2 * NUM_ANCHORS) + NUM_ANCHORS + a];
    float fg = 1.0f / (1.0f + expf(-(s1 - s0)));
    keys[i] = flipKey(fg);
}

// ---------------------------------------------------------------------
// K2: single-workgroup radix-select (WMMA-accelerated digit scan) +
//     stable compaction (+decode only the 6000 survivors) +
//     TDM stage to LDS + 300-round greedy NMS.
// Dynamic LDS layout (bytes):
//   [0,      24064)  smScores : float[PRE_NMS]
//   [24064, 120064)  smBoxes  : float[PRE_NMS*4]   (filled by tensor_load_to_lds)
//   [120064,124160)  rS       : float[1024]
//   [124160,128256)  rI       : int[1024]
//   [128256,129472)  smSel    : int[POST_NMS]
// ---------------------------------------------------------------------
#define DYN_LDS_BYTES 129472

__global__ __launch_bounds__(1024)
void rpn_select_nms_kernel(const float* __restrict__ bb,
                           const float* __restrict__ anchors,
                           const float* __restrict__ imshape,
                           const u32* __restrict__ keys,
                           float* __restrict__ candBoxesG, // ws: PRE_NMS*4 floats
                           float* __restrict__ out)        // POST_NMS*4 floats
{
    const int tid  = threadIdx.x;
    const int B    = 1024;
    const int lane = tid & 31;
    const int wid  = tid >> 5;

    __shared__ uint  hist[256];
    __shared__ float smSuffix[256]; // suffix-summed histogram (from WMMA scan)
    __shared__ float smT[16];       // per-row totals
    __shared__ float smOff[16];     // cross-row suffix offsets
    __shared__ uint  wsum[33];      // per-wave totals -> exclusive offsets, [32]=sum
    __shared__ uint  bc[2];         // radix broadcast: prefix, k
    __shared__ float selBox[4];
    __shared__ int   selValid;

    extern __shared__ char dyn[];
    float* smScores = (float*)(dyn);
    float* smBoxes  = (float*)(dyn + 24064);
    float* rS       = (float*)(dyn + 120064);
    int*   rI       = (int*)  (dyn + 124160);
    int*   smSel    = (int*)  (dyn + 128256);

    for (int j = tid; j < PRE_NMS; j += B) smScores[j] = NEG_INF;

    // ---------- radix select: threshold key T of the 6000th largest ----------
    // Per 8-bit pass: histogram 256 bins, then suffix-sum the bins with
    // V_WMMA_F32_16X16X4_F32:  S1 = H(16x16) x L(16x16),  L[j,c] = (j >= c),
    // done as 4 chained K=4 WMMAs on wave 0 (counts < 2^24 are exact in f32).
    uint prefix = 0, prefMask = 0, k = PRE_NMS;
    for (int pass = 0; pass < 4; ++pass) {
        const int shift = 24 - 8 * pass;
        for (int b = tid; b < 256; b += B) hist[b] = 0;
        __syncthreads();
        for (int i = tid; i < NTOT; i += B) {
            uint u = keys[i];
            if ((u & prefMask) == prefix)
                atomicAdd(&hist[(u >> shift) & 0xFFu], 1u);
        }
        __syncthreads();

        if (tid < 32) { // wave 0, EXEC all ones (WMMA requirement)
            const int hs = lane >> 4;   // lane half: selects K/M offset
            const int m  = lane & 15;   // A row (M) == D column (N) index
            v8f acc = (v8f){0.f, 0.f, 0.f, 0.f, 0.f, 0.f, 0.f, 0.f};
#pragma unroll
            for (int kk = 0; kk < 4; ++kk) {
                const int kbase = kk * 4 + 2 * hs;
                v2f a, b;
                // A 16x4 slice: VGPR0 = K=kbase, VGPR1 = K=kbase+1 (per lane half)
                a.x = (float)hist[m * 16 + kbase + 0];
                a.y = (float)hist[m * 16 + kbase + 1];
                // B 4x16 slice of L: L[j, c] = (j >= c), c = N = m
                b.x = (kbase + 0 >= m) ? 1.0f : 0.0f;
                b.y = (kbase + 1 >= m) ? 1.0f : 0.0f;
                acc = __builtin_amdgcn_wmma_f32_16x16x4_f32(
                    false, a, false, b, (short)0, acc, false, false);
            }
            // acc[v] = S1[M, N] with M = v + 8*hs, N = m  (within-row suffix)
            if (m == 0) {
#pragma unroll
                for (int v = 0; v < 8; ++v) smT[v + 8 * hs] = acc[v]; // row totals
            }
            if (lane == 0) { // 16-element cross-row suffix offsets
                float run = 0.0f;
                for (int r = 15; r >= 0; --r) { smOff[r] = run; run += smT[r]; }
            }
#pragma unroll
            for (int v = 0; v < 8; ++v) {
                const int row = v + 8 * hs;
                smSuffix[row * 16 + m] = acc[v] + smOff[row];
            }
        }
        __syncthreads();

        // unique bin b with  suffix[b] >= k > suffix[b+1]  (suffix is monotone)
        if (tid < 256) {
            const float kf = (float)k;
            float sb = smSuffix[tid];
            float sn = (tid == 255) ? 0.0f : smSuffix[tid + 1];
            if (sb >= kf && sn < kf) {
                bc[0] = prefix | ((uint)tid << shift);
                bc[1] = (uint)(kf - sn);
            }
        }
        __syncthreads();
        prefix = bc[0];
        k      = bc[1];
        prefMask |= (0xFFu << shift);
        __syncthreads();
    }
    const uint T          = prefix;          // exact 32-bit threshold key
    const uint kTies      = k;               // # of keys == T to keep
    const uint cntGreater = PRE_NMS - kTies; // # of keys  > T

    const float ylim = imshape[0];
    const float xlim = imshape[1];

    // ---------- deterministic stable compaction; decode only survivors ------
    for (int phase = 0; phase < 2; ++phase) {
        const uint baseP  = (phase == 0) ? 0u : cntGreater;
        const uint needed = (phase == 0) ? cntGreater : kTies;
        uint base = 0;
        for (int c0 = 0; c0 < NTOT && base < needed; c0 += B) {
            int  i    = c0 + tid;
            uint key  = (i < NTOT) ? keys[i] : 0u;
            uint flag = (i < NTOT) && ((phase == 0) ? (key > T) : (key == T));

            // wave32 ballot scan (zero barriers inside a wave)
            unsigned long long m = __ballot((int)flag);
            uint mm       = (uint)m;
            uint laneRank = __popc(mm & ((1u << lane) - 1u));
            uint wTot     = __popc(mm);
            if (lane == 0) wsum[wid] = wTot;
            __syncthreads();
            if (tid == 0) {
                uint acc = 0;
                for (int w = 0; w < 32; ++w) { uint t = wsum[w]; wsum[w] = acc; acc += t; }
                wsum[32] = acc;
            }
            __syncthreads();
            if (flag) {
                uint slot = baseP + base + wsum[wid] + laneRank;
                if (slot < baseP + needed) {
                    // decode + clip box i (TARGET_STDS=1, TARGET_MEANS=0)
                    float d0 = bb[i * 4 + 0], d1 = bb[i * 4 + 1];
                    float d2 = bb[i * 4 + 2], d3 = bb[i * 4 + 3];
                    float a0 = anchors[i * 4 + 0], a1 = anchors[i * 4 + 1];
                    float a2 = anchors[i * 4 + 2], a3 = anchors[i * 4 + 3];
                    float ah = a2 - a0, aw = a3 - a1;
                    float acy = a0 + 0.5f * ah, acx = a1 + 0.5f * aw;
                    float cy = d0 * ah + acy,  cx = d1 * aw + acx;
                    float h  = ah * expf(d2),  w  = aw * expf(d3);
                    float y1 = fminf(fmaxf(cy - 0.5f * h, 0.0f), ylim);
                    float x1 = fminf(fmaxf(cx - 0.5f * w, 0.0f), xlim);
                    float y2 = fminf(fmaxf(cy + 0.5f * h, 0.0f), ylim);
                    float x2 = fminf(fmaxf(cx + 0.5f * w, 0.0f), xlim);
                    candBoxesG[slot * 4 + 0] = y1;
                    candBoxesG[slot * 4 + 1] = x1;
                    candBoxesG[slot * 4 + 2] = y2;
                    candBoxesG[slot * 4 + 3] = x2;
                    smScores[slot] = unflipKey(key);
                }
            }
            uint tot = wsum[32];
            __syncthreads();
            base += tot;
        }
        __syncthreads();
    }

    // make candidate-box stores visible to the Tensor Data Mover
    __threadfence();
    __syncthreads();

    // ---------- TDM: stage 96 KB candidate-box tile Global -> LDS -----------
    if (tid < 32) { // one wave issues one TDM op
        uint     ldsAddr = (uint)(uintptr_t)(void*)smBoxes; // flat LDS addr[31:0] == LDS offset
        uint64_t ga      = (uint64_t)(uintptr_t)candBoxesG;
        const uint NE = PRE_NMS * 4; // 24000 f32 elements, one row
        // D# group0: count=1, lds_addr, global_addr[56:0], type=2 ("image")
        v4u g0 = (v4u){ 1u,
                        ldsAddr,
                        (uint)(ga & 0xFFFFFFFFu),
                        (uint)((ga >> 32) & 0x01FFFFFFu) | (2u << 30) };
        // D# group1: data_size=4B; tensor_dim0=tile_dim0=24000; dim1=1; strides=24000
        v8i g1 = (v8i){ (int)(2u << 16),               // [17:16] data_size = 2 (4 bytes)
                        (int)((NE & 0xFFFFu) << 16),   // tensor_dim0[15:0]  @ bits 63:48
                        (int)(((NE >> 16) & 0xFFFFu) | (1u << 16)), // dim0[31:16] | tensor_dim1=1
                        (int)((NE & 0xFFFFu) << 16),   // tile_dim0 = 24000  @ bits 127:112
                        (int)1,                        // tile_dim1 = 1
                        (int)NE,                       // tensor_dim0_stride[31:0]
                        (int)((NE & 0xFFFFu) << 16),   // stride0[47:32]=0 | stride1[15:0]
                        (int)(NE >> 16) };             // tensor_dim1_stride[47:16]
        v4i gz4 = (v4i){0, 0, 0, 0};
        v8i gz8 = (v8i){0, 0, 0, 0, 0, 0, 0, 0};
        // clang-23 / therock-10.0 form: 6 args (g0, g1, g2, g3, extra, cpol)
        __builtin_amdgcn_tensor_load_to_lds(g0, g1, gz4, gz4, gz8, 0);
        __builtin_amdgcn_s_wait_tensorcnt(0);
    }
    __syncthreads();

    // ---------- greedy NMS: 300 serial rounds, all state in LDS -------------
    for (int it = 0; it < POST_NMS; ++it) {
        // block-wide argmax (score desc, index asc on ties)
        float best = NEG_INF;
        int   bi   = PRE_NMS; // sentinel
        for (int j = tid; j < PRE_NMS; j += B) {
            float s = smScores[j];
            if (s > best) { best = s; bi = j; }
        }
        rS[tid] = best;
        rI[tid] = bi;
        __syncthreads();
        for (int off = B >> 1; off > 0; off >>= 1) {
            if (tid < off) {
                float s2 = rS[tid + off]; int i2 = rI[tid + off];
                float s1 = rS[tid];       int i1 = rI[tid];
                if (s2 > s1 || (s2 == s1 && i2 < i1)) { rS[tid] = s2; rI[tid] = i2; }
            }
            __syncthreads();
        }
        if (tid == 0) {
            float bs = rS[0];
            int   j  = rI[0];
            int   v  = (bs > NEG_INF) ? 1 : 0;
            smSel[it] = v ? j : -1;
            selValid  = v;
            if (v) {
                selBox[0] = smBoxes[j * 4 + 0];
                selBox[1] = smBoxes[j * 4 + 1];
                selBox[2] = smBoxes[j * 4 + 2];
                selBox[3] = smBoxes[j * 4 + 3];
            }
        }
        __syncthreads();
        if (selValid) {
            float by1 = selBox[0], bx1 = selBox[1], by2 = selBox[2], bx2 = selBox[3];
            float ba = (by2 - by1) * (bx2 - bx1);
            for (int j = tid; j < PRE_NMS; j += B) {
                float y1 = smBoxes[j * 4 + 0], x1 = smBoxes[j * 4 + 1];
                float y2 = smBoxes[j * 4 + 2], x2 = smBoxes[j * 4 + 3];
                float yy1 = fmaxf(by1, y1), xx1 = fmaxf(bx1, x1);
                float yy2 = fminf(by2, y2), xx2 = fminf(bx2, x2);
                float inter = fmaxf(yy2 - yy1, 0.0f) * fmaxf(xx2 - xx1, 0.0f);
                float area  = (y2 - y1) * (x2 - x1);
                float iou   = inter / fmaxf(area + ba - inter, 1e-8f);
                if (iou > IOU_THR) smScores[j] = NEG_INF; // suppress (incl. itself)
            }
        }
        __syncthreads();
    }

    // ---------- emit 300 boxes (zeros where no selection) --------------------
    for (int i = tid; i < POST_NMS; i += B) {
        int s = smSel[i];
        if (s >= 0) {
            out[i * 4 + 0] = smBoxes[s * 4 + 0];
            out[i * 4 + 1] = smBoxes[s * 4 + 1];
            out[i * 4 + 2] = smBoxes[s * 4 + 2];
            out[i * 4 + 3] = smBoxes[s * 4 + 3];
        } else {
            out[i * 4 + 0] = 0.0f;
            out[i * 4 + 1] = 0.0f;
            out[i * 4 + 2] = 0.0f;
            out[i * 4 + 3] = 0.0f;
        }
    }
}

// ---------------------------------------------------------------------
extern "C" void kernel_launch(void* const* d_in, const int* in_sizes, int n_in,
                              void* d_out, int out_size, void* d_ws, size_t ws_size,
                              hipStream_t stream) {
    const float* bb      = (const float*)d_in[0]; // (N,4) deltas
    const float* anchors = (const float*)d_in[1]; // (N,4)
    const float* scores  = (const float*)d_in[2]; // (HW, 18)
    const float* imshape = (const float*)d_in[3]; // (2,)
    float*       out     = (float*)d_out;         // (300,4)

    u32*   keys      = (u32*)d_ws;                                   // N * 4 B
    float* candBoxes = (float*)((char*)d_ws + (size_t)NTOT * sizeof(u32)); // 96 KB

    rpn_score_kernel<<<(NTOT + 255) / 256, 256, 0, stream>>>(scores, keys);
    rpn_select_nms_kernel<<<1, 1024, DYN_LDS_BYTES, stream>>>(
        bb, anchors, imshape, keys, candBoxes, out);
}